// ColorGrid_52673478918226
// MI455X (gfx1250) — compile-verified
//
#include <hip/hip_runtime.h>

#define TEXW 400
#define TEXH 400
#define NTEX (TEXW * TEXH)

typedef float v2f __attribute__((ext_vector_type(2)));
typedef float v4f __attribute__((ext_vector_type(4)));

// ---------------------------------------------------------------------------
// Pass 1: repack planar [3][400][400] color + [3][400][400] grid into an
// interleaved, 32B-padded texel array: tex[y*400+x] = {c0,c1,c2,g0,g1,g2,0,0}.
// 5 MB total -> fully resident in the MI455X's 192 MB L2. Turns 24 scattered
// scalar gathers per sample point into 8 wide vector loads.
// ---------------------------------------------------------------------------
__global__ void ColorGrid_repack_kernel(const float* __restrict__ color,
                                        const float* __restrict__ grid,
                                        float* __restrict__ tex) {
    int t = blockIdx.x * blockDim.x + threadIdx.x;
    if (t >= NTEX) return;
    v4f a, b;
    a.x = color[t];
    a.y = color[t + NTEX];
    a.z = color[t + 2 * NTEX];
    a.w = grid[t];
    b.x = grid[t + NTEX];
    b.y = grid[t + 2 * NTEX];
    b.z = 0.0f;
    b.w = 0.0f;
    v4f* dst = (v4f*)(tex + (size_t)t * 8);
    dst[0] = a;
    dst[1] = b;
}

// ---------------------------------------------------------------------------
// One sample point: bilinear 4-corner gather from packed texels + sigmoid.
// Returns 6 floats through out[0..5] = {sig(c0..c2), g0..g2}.
// ---------------------------------------------------------------------------
__device__ __forceinline__ void sample_point(const float* __restrict__ tex,
                                             float px, float py,
                                             float* __restrict__ r) {
    // Replicate reference op order: coords = x*2-1 ; ix = ((coords+1)*W-1)*0.5
    float cx = px * 2.0f - 1.0f;
    float cy = py * 2.0f - 1.0f;
    float ix = ((cx + 1.0f) * (float)TEXW - 1.0f) * 0.5f;
    float iy = ((cy + 1.0f) * (float)TEXH - 1.0f) * 0.5f;

    float x0f = floorf(ix), y0f = floorf(iy);
    float fx = ix - x0f, fy = iy - y0f;   // wx1, wy1
    float gx = 1.0f - fx, gy = 1.0f - fy; // wx0, wy0

    int x0 = (int)x0f, y0 = (int)y0f;
    int x1 = x0 + 1,   y1 = y0 + 1;

    // Per-corner validity folded into weights (padding_mode='zeros')
    float vx0 = (x0 >= 0 && x0 <= TEXW - 1) ? 1.0f : 0.0f;
    float vx1 = (x1 >= 0 && x1 <= TEXW - 1) ? 1.0f : 0.0f;
    float vy0 = (y0 >= 0 && y0 <= TEXH - 1) ? 1.0f : 0.0f;
    float vy1 = (y1 >= 0 && y1 <= TEXH - 1) ? 1.0f : 0.0f;

    float w00 = gx * gy * vx0 * vy0;
    float w10 = fx * gy * vx1 * vy0;
    float w01 = gx * fy * vx0 * vy1;
    float w11 = fx * fy * vx1 * vy1;

    int cx0 = min(max(x0, 0), TEXW - 1);
    int cx1 = min(max(x1, 0), TEXW - 1);
    int cy0 = min(max(y0, 0), TEXH - 1);
    int cy1 = min(max(y1, 0), TEXH - 1);

    float a0 = 0.f, a1 = 0.f, a2 = 0.f, a3 = 0.f, a4 = 0.f, a5 = 0.f;

    #define CORNER(CY, CX, W)                                              \
    {                                                                      \
        const float* p = tex + ((size_t)(CY) * TEXW + (CX)) * 8;           \
        v4f ta = *(const v4f*)p;          /* c0 c1 c2 g0 : b128 */         \
        v2f tb = *(const v2f*)(p + 4);    /* g1 g2       : b64  */         \
        a0 = fmaf((W), ta.x, a0);                                          \
        a1 = fmaf((W), ta.y, a1);                                          \
        a2 = fmaf((W), ta.z, a2);                                          \
        a3 = fmaf((W), ta.w, a3);                                          \
        a4 = fmaf((W), tb.x, a4);                                          \
        a5 = fmaf((W), tb.y, a5);                                          \
    }
    CORNER(cy0, cx0, w00)
    CORNER(cy0, cx1, w10)
    CORNER(cy1, cx0, w01)
    CORNER(cy1, cx1, w11)
    #undef CORNER

    r[0] = 1.0f / (1.0f + expf(-a0));
    r[1] = 1.0f / (1.0f + expf(-a1));
    r[2] = 1.0f / (1.0f + expf(-a2));
    r[3] = a3;
    r[4] = a4;
    r[5] = a5;
}

// ---------------------------------------------------------------------------
// Pass 2: TWO points per thread. One b128 coord load (2 x float2), 16 in-
// flight corner gathers for latency hiding, and the 12 output floats land at
// a 16B-aligned 48*j offset -> exactly three nontemporal global_store_b128.
// ---------------------------------------------------------------------------
__global__ void __launch_bounds__(256)
ColorGrid_sample_kernel(const float* __restrict__ x,
                        const float* __restrict__ tex,
                        float* __restrict__ out,
                        int npairs) {
    int j = blockIdx.x * blockDim.x + threadIdx.x;
    if (j >= npairs) return;

    // gfx1250 global_prefetch_b8 on the coalesced coordinate stream.
    if (j + 2048 < npairs) {
        __builtin_prefetch(x + 4 * (size_t)(j + 2048), 0, 3);
    }

    v4f c = *(const v4f*)(x + 4 * (size_t)j);   // {x0,y0,x1,y1}

    float r0[6], r1[6];
    sample_point(tex, c.x, c.y, r0);
    sample_point(tex, c.z, c.w, r1);

    // Streaming NT stores (th:TH_STORE_NT): write-once 100MB output must not
    // evict the L2-resident texel table / coord stream across graph replays.
    float* o = out + (size_t)j * 12;
    v4f o0 = {r0[0], r0[1], r0[2], r0[3]};
    v4f o1 = {r0[4], r0[5], r1[0], r1[1]};
    v4f o2 = {r1[2], r1[3], r1[4], r1[5]};
    __builtin_nontemporal_store(o0, (v4f*)(o + 0));
    __builtin_nontemporal_store(o1, (v4f*)(o + 4));
    __builtin_nontemporal_store(o2, (v4f*)(o + 8));
}

// Tail kernel for a possible odd final point (npts odd; not hit for 4194304).
__global__ void ColorGrid_tail_kernel(const float* __restrict__ x,
                                      const float* __restrict__ tex,
                                      float* __restrict__ out,
                                      int ptidx) {
    if (blockIdx.x != 0 || threadIdx.x != 0) return;
    float r[6];
    sample_point(tex, x[2 * (size_t)ptidx], x[2 * (size_t)ptidx + 1], r);
    float* o = out + (size_t)ptidx * 6;
    #pragma unroll
    for (int k = 0; k < 6; ++k) o[k] = r[k];
}

// ---------------------------------------------------------------------------
// Fallback: direct gather from planar layouts (only if workspace too small).
// ---------------------------------------------------------------------------
__global__ void __launch_bounds__(256)
ColorGrid_direct_kernel(const float* __restrict__ x,
                        const float* __restrict__ color,
                        const float* __restrict__ grid,
                        float* __restrict__ out,
                        int npts) {
    int i = blockIdx.x * blockDim.x + threadIdx.x;
    if (i >= npts) return;

    v2f c = *(const v2f*)(x + 2 * (size_t)i);
    float cx = c.x * 2.0f - 1.0f;
    float cy = c.y * 2.0f - 1.0f;
    float ix = ((cx + 1.0f) * (float)TEXW - 1.0f) * 0.5f;
    float iy = ((cy + 1.0f) * (float)TEXH - 1.0f) * 0.5f;
    float x0f = floorf(ix), y0f = floorf(iy);
    float fx = ix - x0f, fy = iy - y0f;
    float gx = 1.0f - fx, gy = 1.0f - fy;
    int x0 = (int)x0f, y0 = (int)y0f;
    int x1 = x0 + 1,   y1 = y0 + 1;
    float vx0 = (x0 >= 0 && x0 <= TEXW - 1) ? 1.0f : 0.0f;
    float vx1 = (x1 >= 0 && x1 <= TEXW - 1) ? 1.0f : 0.0f;
    float vy0 = (y0 >= 0 && y0 <= TEXH - 1) ? 1.0f : 0.0f;
    float vy1 = (y1 >= 0 && y1 <= TEXH - 1) ? 1.0f : 0.0f;
    float w00 = gx * gy * vx0 * vy0;
    float w10 = fx * gy * vx1 * vy0;
    float w01 = gx * fy * vx0 * vy1;
    float w11 = fx * fy * vx1 * vy1;
    int cx0 = min(max(x0, 0), TEXW - 1);
    int cx1 = min(max(x1, 0), TEXW - 1);
    int cy0 = min(max(y0, 0), TEXH - 1);
    int cy1 = min(max(y1, 0), TEXH - 1);
    int i00 = cy0 * TEXW + cx0, i10 = cy0 * TEXW + cx1;
    int i01 = cy1 * TEXW + cx0, i11 = cy1 * TEXW + cx1;

    float o[6];
    #pragma unroll
    for (int ch = 0; ch < 3; ++ch) {
        const float* cp = color + (size_t)ch * NTEX;
        const float* gp = grid + (size_t)ch * NTEX;
        float cv = w00 * cp[i00] + w10 * cp[i10] + w01 * cp[i01] + w11 * cp[i11];
        float gv = w00 * gp[i00] + w10 * gp[i10] + w01 * gp[i01] + w11 * gp[i11];
        o[ch] = 1.0f / (1.0f + expf(-cv));
        o[ch + 3] = gv;
    }
    float* op = out + (size_t)i * 6;
    v2f o0 = {o[0], o[1]}, o1 = {o[2], o[3]}, o2 = {o[4], o[5]};
    __builtin_nontemporal_store(o0, (v2f*)(op + 0));
    __builtin_nontemporal_store(o1, (v2f*)(op + 2));
    __builtin_nontemporal_store(o2, (v2f*)(op + 4));
}

extern "C" void kernel_launch(void* const* d_in, const int* in_sizes, int n_in,
                              void* d_out, int out_size, void* d_ws, size_t ws_size,
                              hipStream_t stream) {
    const float* x     = (const float*)d_in[0];   // [8, 524288, 2]
    const float* color = (const float*)d_in[1];   // [1, 3, 400, 400]
    const float* grid  = (const float*)d_in[2];   // [1, 3, 400, 400]
    float* out = (float*)d_out;                   // [8, 524288, 6]

    const int npts = in_sizes[0] / 2;             // 4,194,304
    const size_t tex_bytes = (size_t)NTEX * 8 * sizeof(float);  // 5.12 MB

    const int BLK = 256;
    if (ws_size >= tex_bytes) {
        float* tex = (float*)d_ws;
        ColorGrid_repack_kernel<<<(NTEX + BLK - 1) / BLK, BLK, 0, stream>>>(
            color, grid, tex);
        const int npairs = npts / 2;
        ColorGrid_sample_kernel<<<(npairs + BLK - 1) / BLK, BLK, 0, stream>>>(
            x, tex, out, npairs);
        if (npts & 1) {
            ColorGrid_tail_kernel<<<1, 32, 0, stream>>>(x, tex, out, npts - 1);
        }
    } else {
        ColorGrid_direct_kernel<<<(npts + BLK - 1) / BLK, BLK, 0, stream>>>(
            x, color, grid, out, npts);
    }
}